// GenView_32985348833477
// MI455X (gfx1250) — compile-verified
//
#include <hip/hip_runtime.h>
#include <hip/hip_bf16.h>
#include <math.h>

#define N_NODES 50000
#define N_EDGES 800000
#define F_IN    512
#define HID     128

static_assert(N_NODES % 16 == 0, "row tiling assumes N divisible by 16");
static_assert(F_IN % 4 == 0, "K loop assumes F_IN divisible by 4");

typedef __attribute__((ext_vector_type(2))) float v2f;
typedef __attribute__((ext_vector_type(8))) float v8f;

// ---- order-preserving float <-> uint encoding for atomicMax on floats ----
__device__ __forceinline__ unsigned ord_encode(float f) {
  unsigned u = __float_as_uint(f);
  return (u & 0x80000000u) ? ~u : (u | 0x80000000u);
}
__device__ __forceinline__ float ord_decode(unsigned u) {
  return (u & 0x80000000u) ? __uint_as_float(u & 0x7FFFFFFFu)
                           : __uint_as_float(~u);
}

// ---------------- init: zero embsum [N*HID], ssum [N], mord [N] ----------------
__global__ __launch_bounds__(256) void k_init(float* embsum, float* ssum, unsigned* mord) {
  int i = blockIdx.x * 256 + threadIdx.x;
  if (i < N_NODES * HID) embsum[i] = 0.0f;
  if (i < N_NODES) { ssum[i] = 0.0f; mord[i] = 0u; }   // 0 < encode(-inf)=0x007FFFFF
}

// ---------------- GEMM: support = feat @ gcn_w via V_WMMA_F32_16X16X4_F32 ------
// Block = 256 threads = 8 waves. Block computes rows [16*bid, 16*bid+16) x all 128 cols.
// Wave w owns the 16-column tile [16w, 16w+16). K-loop: 512/4 = 128 wmma ops.
// N_NODES % 16 == 0 -> no row guards anywhere (clean store epilogue).
#define APITCH 517   // odd pitch -> 16 rows hit 16 distinct LDS banks (64 banks)
__global__ __launch_bounds__(256) void k_gemm_wmma(const float* __restrict__ feat,
                                                   const float* __restrict__ w,
                                                   float* __restrict__ support) {
  __shared__ float At[16 * APITCH];
  const int rowBase = blockIdx.x * 16;
  const int t = threadIdx.x;

  // stage a 16 x 512 tile of feat into LDS (coalesced, unguarded)
  for (int i = t; i < 16 * F_IN; i += 256) {
    int r = i >> 9;          // /512
    int c = i & (F_IN - 1);
    At[r * APITCH + c] = feat[(size_t)(rowBase + r) * F_IN + c];
  }
  __syncthreads();

  const int wave = t >> 5;           // 0..7 -> column tile
  const int lane = t & 31;
  const int m    = lane & 15;        // A row within tile
  const int kho  = (lane >> 4) * 2;  // lanes 16-31 hold K=2,3 (ISA 7.12.2)
  const int n    = wave * 16 + (lane & 15);  // B/D column

  v8f acc = {};
  for (int k = 0; k < F_IN; k += 4) {
    v2f a, b;
    a.x = At[m * APITCH + k + kho];          // VGPR0: K = kho
    a.y = At[m * APITCH + k + kho + 1];      // VGPR1: K = kho+1
    b.x = w[(size_t)(k + kho) * HID + n];    // B row K=kho striped over lanes
    b.y = w[(size_t)(k + kho + 1) * HID + n];
    acc = __builtin_amdgcn_wmma_f32_16x16x4_f32(
        /*neg_a=*/false, a, /*neg_b=*/false, b,
        /*c_mod=*/(short)0, acc, /*reuse_a=*/false, /*reuse_b=*/false);
  }

  // C/D layout: VGPR i -> row (i) for lanes 0-15, row (i+8) for lanes 16-31; col = lane&15
  const int rowOut = rowBase + (lane >> 4) * 8;
#pragma unroll
  for (int i = 0; i < 8; ++i) {
    support[(size_t)(rowOut + i) * HID + n] = acc[i];
  }
}

// ---------------- edge scatter: embsum[row] += v * support[col] ----------------
// one wave per edge; lane handles 4 consecutive features (float4 gather, 4 atomics)
__global__ __launch_bounds__(256) void k_scatter(const float* __restrict__ vvals,
                                                 const long long* __restrict__ idx,
                                                 const float* __restrict__ support,
                                                 float* __restrict__ embsum) {
  long long gid = (long long)blockIdx.x * 256 + threadIdx.x;
  int e = (int)(gid >> 5);
  if (e >= N_EDGES) return;
  int lane = threadIdx.x & 31;
  int r = (int)idx[e];
  int c = (int)idx[N_EDGES + e];
  float v = vvals[e];
  float4 sv = ((const float4*)(support + (size_t)c * HID))[lane];
  float* dst = embsum + (size_t)r * HID + lane * 4;
  atomicAdd(dst + 0, v * sv.x);
  atomicAdd(dst + 1, v * sv.y);
  atomicAdd(dst + 2, v * sv.z);
  atomicAdd(dst + 3, v * sv.w);
}

// ---------------- per-node scores: s1 = (emb+b).w1, s2 = (emb+b).w2 ------------
// MLP is linear over concat: dot(concat(a,b),w) = a.w1 + b.w2 -> per-node precompute.
__global__ __launch_bounds__(256) void k_node_scores(const float* __restrict__ embsum,
                                                     const float* __restrict__ gcn_b,
                                                     const float* __restrict__ mlp_w,
                                                     float* __restrict__ s1,
                                                     float* __restrict__ s2) {
  int node = blockIdx.x * 8 + (threadIdx.x >> 5);
  if (node >= N_NODES) return;
  int lane = threadIdx.x & 31;
  float4 eb = ((const float4*)(embsum + (size_t)node * HID))[lane];
  float4 bb = ((const float4*)gcn_b)[lane];
  float4 w1 = ((const float4*)mlp_w)[lane];
  float4 w2 = ((const float4*)(mlp_w + HID))[lane];
  float e0 = eb.x + bb.x, e1 = eb.y + bb.y, e2 = eb.z + bb.z, e3 = eb.w + bb.w;
  float d1 = e0 * w1.x + e1 * w1.y + e2 * w1.z + e3 * w1.w;
  float d2 = e0 * w2.x + e1 * w2.y + e2 * w2.z + e3 * w2.w;
#pragma unroll
  for (int off = 16; off > 0; off >>= 1) {
    d1 += __shfl_xor(d1, off, 32);
    d2 += __shfl_xor(d2, off, 32);
  }
  if (lane == 0) { s1[node] = d1; s2[node] = d2; }
}

// ---------------- per-edge temp + segment max (ordered-uint atomicMax) ---------
__global__ __launch_bounds__(256) void k_edge_temp_max(const long long* __restrict__ idx,
                                                       const float* __restrict__ s1,
                                                       const float* __restrict__ s2,
                                                       const float* __restrict__ mlp_b,
                                                       float* __restrict__ temp,
                                                       unsigned* __restrict__ mord) {
  int e = blockIdx.x * 256 + threadIdx.x;
  if (e >= N_EDGES) return;
  int r = (int)idx[e];
  int c = (int)idx[N_EDGES + e];
  float tval = s1[r] + s2[c] + mlp_b[0];
  temp[e] = tval;
  atomicMax(&mord[r], ord_encode(tval));
}

// ---------------- per-edge exp + segment sum -----------------------------------
__global__ __launch_bounds__(256) void k_edge_exp_sum(const long long* __restrict__ idx,
                                                      const float* __restrict__ temp,
                                                      const unsigned* __restrict__ mord,
                                                      float* __restrict__ ews,
                                                      float* __restrict__ ssum) {
  int e = blockIdx.x * 256 + threadIdx.x;
  if (e >= N_EDGES) return;
  int r = (int)idx[e];
  float ex = expf(temp[e] - ord_decode(mord[r]));
  ews[e] = ex;
  atomicAdd(&ssum[r], ex);
}

// ---------------- final: out = v_ori + pi --------------------------------------
__global__ __launch_bounds__(256) void k_final(const float* __restrict__ vvals,
                                               const long long* __restrict__ idx,
                                               const float* __restrict__ ews,
                                               const float* __restrict__ ssum,
                                               float* __restrict__ out) {
  int e = blockIdx.x * 256 + threadIdx.x;
  if (e >= N_EDGES) return;
  int r = (int)idx[e];
  out[e] = vvals[e] + ews[e] / ssum[r];   // COM_LAMBDA = 1
}

extern "C" void kernel_launch(void* const* d_in, const int* in_sizes, int n_in,
                              void* d_out, int out_size, void* d_ws, size_t ws_size,
                              hipStream_t stream) {
  const float*     v_ori = (const float*)d_in[0];
  const float*     feat  = (const float*)d_in[1];
  const long long* idx   = (const long long*)d_in[2];  // int64 [2, E]
  // d_in[3] = num_node (compile-time constant N_NODES)
  const float*     gcn_w = (const float*)d_in[4];
  const float*     gcn_b = (const float*)d_in[5];
  const float*     mlp_w = (const float*)d_in[6];      // [1, 2*HID]
  const float*     mlp_b = (const float*)d_in[7];
  float*           out   = (float*)d_out;

  // workspace layout (all counts divisible by 4 -> 16B aligned)
  float* ws = (float*)d_ws;
  float*    support = ws;                               // N*HID
  float*    embsum  = support + (size_t)N_NODES * HID;  // N*HID
  float*    s1      = embsum + (size_t)N_NODES * HID;   // N
  float*    s2      = s1 + N_NODES;                     // N
  float*    temp    = s2 + N_NODES;                     // E
  float*    ews     = temp + N_EDGES;                   // E
  float*    ssum    = ews + N_EDGES;                    // N
  unsigned* mord    = (unsigned*)(ssum + N_NODES);      // N

  const int initN = N_NODES * HID;
  k_init<<<(initN + 255) / 256, 256, 0, stream>>>(embsum, ssum, mord);

  k_gemm_wmma<<<N_NODES / 16, 256, 0, stream>>>(feat, gcn_w, support);

  long long scatterT = (long long)N_EDGES * 32;
  k_scatter<<<(unsigned)((scatterT + 255) / 256), 256, 0, stream>>>(v_ori, idx, support, embsum);

  k_node_scores<<<(N_NODES + 7) / 8, 256, 0, stream>>>(embsum, gcn_b, mlp_w, s1, s2);

  int eblocks = (N_EDGES + 255) / 256;
  k_edge_temp_max<<<eblocks, 256, 0, stream>>>(idx, s1, s2, mlp_b, temp, mord);
  k_edge_exp_sum<<<eblocks, 256, 0, stream>>>(idx, temp, mord, ews, ssum);
  k_final<<<eblocks, 256, 0, stream>>>(v_ori, idx, ews, ssum, out);
}